// EndoMamba_14894946583159
// MI455X (gfx1250) — compile-verified
//
#include <hip/hip_runtime.h>
#include <math.h>

typedef __attribute__((ext_vector_type(2))) float v2f;
typedef __attribute__((ext_vector_type(8))) float v8f;

// ---------------- problem constants ----------------
#define BB   2
#define TT   4
#define NP   196          // 14*14 patches
#define LL   784          // T*N tokens per batch
#define MM   1568         // B*L rows
#define EE   384
#define DI   768
#define SS   8
#define RR   24
#define XD   40           // R + 2S
#define F2   1536         // 2*Di
#define KP   768          // patch K = C*P*P
#define DEPTH 12
#define NSPA  6

// =====================================================================
// TN GEMM via V_WMMA_F32_16X16X4_F32, register-blocked:
//   C[m,n] = sum_k A[m*lda+k] * B[n*ldb+k]
// Block = 128 threads (4 waves, 2x2) -> block tile 64x64.
// Wave tile 32x32 = 4 independent accumulators; per k-step load
// 2 A frags + 2 B frags and issue 4 WMMAs (8 FLOP/byte, 4 indep chains).
// Lane layout per CDNA5 ISA 7.12.2:
//   A 16x4 : lane r(0..15) v0=A[r][k+2h], v1=A[r][k+2h+1]  (h = lane>>4)
//   B 4x16 : lane r        v0=B[k+2h][r], v1=B[k+2h+1][r]
//   D 16x16: element g of lane -> D[g + 8h][r]
// =====================================================================
__global__ __launch_bounds__(128)
void wmma_gemm_tn(const float* __restrict__ A, int lda,
                  const float* __restrict__ B, int ldb,
                  float* __restrict__ C, int ldc,
                  int M, int N, int K)
{
    const int wave = threadIdx.x >> 5;
    const int lane = threadIdx.x & 31;
    const int half = lane >> 4;
    const int r    = lane & 15;

    const int m0 = blockIdx.y * 64 + (wave >> 1) * 32;
    const int n0 = blockIdx.x * 64 + (wave & 1) * 32;

    // clamp loads to valid rows; stores are predicated
    const int mr0 = min(m0 + r,      M - 1);
    const int mr1 = min(m0 + 16 + r, M - 1);
    const int nr0 = min(n0 + r,      N - 1);
    const int nr1 = min(n0 + 16 + r, N - 1);

    const float* __restrict__ A0 = A + (size_t)mr0 * lda + 2 * half;
    const float* __restrict__ A1 = A + (size_t)mr1 * lda + 2 * half;
    const float* __restrict__ B0 = B + (size_t)nr0 * ldb + 2 * half;
    const float* __restrict__ B1 = B + (size_t)nr1 * ldb + 2 * half;

    v8f acc00 = {}, acc01 = {}, acc10 = {}, acc11 = {};

#pragma unroll 2
    for (int k = 0; k < K; k += 4) {
        v2f a0 = *(const v2f*)(A0 + k);
        v2f a1 = *(const v2f*)(A1 + k);
        v2f b0 = *(const v2f*)(B0 + k);
        v2f b1 = *(const v2f*)(B1 + k);
        __builtin_prefetch(A0 + k + 64, 0, 0);
        __builtin_prefetch(A1 + k + 64, 0, 0);
        __builtin_prefetch(B0 + k + 64, 0, 0);
        __builtin_prefetch(B1 + k + 64, 0, 0);
        acc00 = __builtin_amdgcn_wmma_f32_16x16x4_f32(false, a0, false, b0, (short)0, acc00, false, false);
        acc01 = __builtin_amdgcn_wmma_f32_16x16x4_f32(false, a0, false, b1, (short)0, acc01, false, false);
        acc10 = __builtin_amdgcn_wmma_f32_16x16x4_f32(false, a1, false, b0, (short)0, acc10, false, false);
        acc11 = __builtin_amdgcn_wmma_f32_16x16x4_f32(false, a1, false, b1, (short)0, acc11, false, false);
    }

    const int nA = n0 + r;
    const int nB = n0 + 16 + r;
#pragma unroll
    for (int g = 0; g < 8; ++g) {
        int mA = m0 + g + 8 * half;
        int mB = m0 + 16 + g + 8 * half;
        if (mA < M && nA < N) C[(size_t)mA * ldc + nA] = acc00[g];
        if (mA < M && nB < N) C[(size_t)mA * ldc + nB] = acc01[g];
        if (mB < M && nA < N) C[(size_t)mB * ldc + nA] = acc10[g];
        if (mB < M && nB < N) C[(size_t)mB * ldc + nB] = acc11[g];
    }
}

// =====================================================================
// Patch embedding: im2col gather  x[B,C,T,H,W] -> Ap[M=1568][K=768]
// Ap[(b*T+t)*196 + hw][c*256 + p*16 + q] = x[b,c,t,hp*16+p,wp*16+q]
// =====================================================================
__global__ void im2col_kernel(const float* __restrict__ x, float* __restrict__ Ap)
{
    int idx = blockIdx.x * 256 + threadIdx.x;
    if (idx >= MM * KP) return;
    int kk = idx % KP, m = idx / KP;
    int c = kk >> 8, pq = kk & 255, p = pq >> 4, q = pq & 15;
    int hw = m % NP, bt = m / NP, t = bt % TT, b = bt / TT;
    int hp = hw / 14, wp = hw % 14;
    Ap[idx] = x[(((size_t)(b * 3 + c) * TT + t) * 224 + hp * 16 + p) * 224 + wp * 16 + q];
}

// h += patch_b[e] + pos_embed[hw,e] + temporal_pe[t,e]
__global__ void embed_add_kernel(float* __restrict__ h,
                                 const float* __restrict__ pb,
                                 const float* __restrict__ pos)
{
    int idx = blockIdx.x * 256 + threadIdx.x;
    if (idx >= MM * EE) return;
    int e = idx % EE, m = idx / EE;
    int hw = m % NP, t = (m / NP) % TT;
    int j = e >> 1;
    float div = expf(-logf(10000.f) * (2.f * j) / (float)EE);
    float ang = (float)t * div;
    float pev = (e & 1) ? cosf(ang) : sinf(ang);
    h[idx] += pb[e] + pos[hw * EE + e] + pev;
}

// =====================================================================
// residual = first ? h : residual + h ;  out = rmsnorm(residual) * w
// One block (128 threads) per token; E = 384 = 3*128.
// =====================================================================
__global__ __launch_bounds__(128)
void resnorm_kernel(const float* __restrict__ h, float* __restrict__ residual,
                    const float* __restrict__ w, float* __restrict__ out, int first)
{
    __shared__ float red[4];
    const int m = blockIdx.x;
    const float* hr = h + (size_t)m * EE;
    float*       rr = residual + (size_t)m * EE;

    float v[3];
    float ss = 0.f;
#pragma unroll
    for (int i = 0; i < 3; ++i) {
        int e = threadIdx.x + i * 128;
        float val = hr[e];
        if (!first) val += rr[e];
        rr[e] = val;
        v[i] = val;
        ss += val * val;
    }
    // wave reduce (wave32)
    ss += __shfl_xor(ss, 16);
    ss += __shfl_xor(ss, 8);
    ss += __shfl_xor(ss, 4);
    ss += __shfl_xor(ss, 2);
    ss += __shfl_xor(ss, 1);
    if ((threadIdx.x & 31) == 0) red[threadIdx.x >> 5] = ss;
    __syncthreads();
    float total = red[0] + red[1] + red[2] + red[3];
    float scale = rsqrtf(total / (float)EE + 1e-5f);
#pragma unroll
    for (int i = 0; i < 3; ++i) {
        int e = threadIdx.x + i * 128;
        out[(size_t)m * EE + e] = v[i] * scale * w[e];
    }
}

// =====================================================================
// Causal depthwise conv1d (K=4) + SiLU.  xm lives in xz[:, 0:768] (ld F2).
// reverse=1: operate on the time-reversed sequence (output in reversed coords).
// =====================================================================
__global__ void conv_silu_kernel(const float* __restrict__ xz,
                                 const float* __restrict__ cw,
                                 const float* __restrict__ cb,
                                 float* __restrict__ xc, int reverse)
{
    int idx = blockIdx.x * 256 + threadIdx.x;
    if (idx >= BB * LL * DI) return;
    int d = idx % DI;
    int l = (idx / DI) % LL;
    int b = idx / (DI * LL);
    float acc = cb[d];
#pragma unroll
    for (int k = 0; k < 4; ++k) {
        int j = l - 3 + k;
        if (j >= 0) {
            int src = reverse ? (LL - 1 - j) : j;
            acc += cw[d * 4 + k] * xz[((size_t)b * LL + src) * F2 + d];
        }
    }
    float sg = 1.f / (1.f + __expf(-acc));
    xc[idx] = acc * sg;
}

// =====================================================================
// Selective scan. One lane per (b, d, s): 12288 lanes, 8-lane groups
// reduce y over S with shfl_xor. Forward writes y; backward accumulates
// into the reversed position.
// =====================================================================
__global__ __launch_bounds__(256)
void scan_kernel(const float* __restrict__ xc,   // [B,L,Di] (dir coords)
                 const float* __restrict__ xdbl, // [B,L,40]
                 const float* __restrict__ dtlin,// [B,L,Di]
                 const float* __restrict__ dtb,  // [Di]
                 const float* __restrict__ A_log,// [Di,S]
                 const float* __restrict__ Dp,   // [Di]
                 float* __restrict__ y,          // [B,L,Di] (original coords)
                 int reverse)
{
    int gid = blockIdx.x * 256 + threadIdx.x;
    int s  = gid & 7;
    int ch = gid >> 3;
    if (ch >= BB * DI) return;
    int d = ch % DI, b = ch / DI;

    const float A    = -__expf(A_log[d * SS + s]);
    const float bias = dtb[d];
    const float Dv   = Dp[d];

    const float* xcb = xc    + (size_t)b * LL * DI + d;
    const float* dtp = dtlin + (size_t)b * LL * DI + d;
    const float* xdb = xdbl  + (size_t)b * LL * XD;
    float*       yb  = y     + (size_t)b * LL * DI + d;

    float hstate = 0.f;
    for (int l = 0; l < LL; ++l) {
        float dtv = dtp[(size_t)l * DI] + bias;
        dtv = (dtv > 20.f) ? dtv : log1pf(__expf(dtv));       // softplus
        float u  = xcb[(size_t)l * DI];
        float Bv = xdb[l * XD + RR + s];
        float Cv = xdb[l * XD + RR + SS + s];
        hstate = __expf(dtv * A) * hstate + dtv * u * Bv;
        float yp = hstate * Cv;
        yp += __shfl_xor(yp, 1);
        yp += __shfl_xor(yp, 2);
        yp += __shfl_xor(yp, 4);
        if (s == 0) {
            float o = yp + u * Dv;
            int lo = reverse ? (LL - 1 - l) : l;
            if (reverse) yb[(size_t)lo * DI] += o;
            else         yb[(size_t)lo * DI]  = o;
        }
    }
}

// y *= silu(z), z = xz[:, 768:1536]  (silu(z) factors out of fwd+bwd sum)
__global__ void gate_kernel(float* __restrict__ y, const float* __restrict__ xz)
{
    int idx = blockIdx.x * 256 + threadIdx.x;
    if (idx >= MM * DI) return;
    int d = idx % DI, m = idx / DI;
    float z  = xz[(size_t)m * F2 + DI + d];
    float sg = z / (1.f + __expf(-z));
    y[idx] *= sg;
}

// =====================================================================
extern "C" void kernel_launch(void* const* d_in, const int* in_sizes, int n_in,
                              void* d_out, int out_size, void* d_ws, size_t ws_size,
                              hipStream_t stream)
{
    const float* x         = (const float*)d_in[0];
    const float* patch_w   = (const float*)d_in[1];   // [E,768]
    const float* patch_b   = (const float*)d_in[2];
    const float* pos_embed = (const float*)d_in[3];   // [N,E]
    const float* in_proj_w = (const float*)d_in[4];   // [depth,1536,E]
    const float* conv_w    = (const float*)d_in[5];   // [depth,Di,4]
    const float* conv_b    = (const float*)d_in[6];
    const float* xproj_w   = (const float*)d_in[7];   // [depth,40,Di]
    const float* dtproj_w  = (const float*)d_in[8];   // [depth,Di,24]
    const float* dtproj_b  = (const float*)d_in[9];
    const float* A_log     = (const float*)d_in[10];  // [depth,Di,8]
    const float* D_param   = (const float*)d_in[11];
    const float* outproj_w = (const float*)d_in[12];  // [depth,E,Di]
    const float* norm_w    = (const float*)d_in[13];  // [depth,E]
    const float* conv_wb   = (const float*)d_in[14];
    const float* conv_bb   = (const float*)d_in[15];
    const float* xproj_wb  = (const float*)d_in[16];
    const float* dtproj_wb = (const float*)d_in[17];
    const float* dtproj_bb = (const float*)d_in[18];
    const float* A_log_b   = (const float*)d_in[19];
    const float* D_b       = (const float*)d_in[20];
    const float* norm_f_w  = (const float*)d_in[21];

    float* ws   = (float*)d_ws;
    float* Ap   = ws;                          // M*768
    float* hbuf = Ap   + (size_t)MM * KP;      // M*E
    float* res  = hbuf + (size_t)MM * EE;      // M*E
    float* hn   = res  + (size_t)MM * EE;      // M*E
    float* xz   = hn   + (size_t)MM * EE;      // M*1536
    float* xc   = xz   + (size_t)MM * F2;      // M*Di
    float* xdbl = xc   + (size_t)MM * DI;      // M*40
    float* dtl  = xdbl + (size_t)MM * XD;      // M*Di
    float* ybuf = dtl  + (size_t)MM * DI;      // M*Di

    const dim3 blk128(128), blk256(256);
    const int gy = (MM + 63) / 64;             // 25 block rows

    // ---- patch embedding ----
    im2col_kernel<<<(MM * KP + 255) / 256, blk256, 0, stream>>>(x, Ap);
    wmma_gemm_tn<<<dim3((EE + 63) / 64, gy), blk128, 0, stream>>>(
        Ap, KP, patch_w, KP, hbuf, EE, MM, EE, KP);
    embed_add_kernel<<<(MM * EE + 255) / 256, blk256, 0, stream>>>(hbuf, patch_b, pos_embed);

    // ---- layers ----
    for (int i = 0; i < DEPTH; ++i) {
        resnorm_kernel<<<MM, blk128, 0, stream>>>(hbuf, res, norm_w + (size_t)i * EE, hn, i == 0);

        wmma_gemm_tn<<<dim3((F2 + 63) / 64, gy), blk128, 0, stream>>>(
            hn, EE, in_proj_w + (size_t)i * F2 * EE, EE, xz, F2, MM, F2, EE);

        // forward direction
        conv_silu_kernel<<<(MM * DI + 255) / 256, blk256, 0, stream>>>(
            xz, conv_w + (size_t)i * DI * 4, conv_b + (size_t)i * DI, xc, 0);
        wmma_gemm_tn<<<dim3(1, gy), blk128, 0, stream>>>(
            xc, DI, xproj_w + (size_t)i * XD * DI, DI, xdbl, XD, MM, XD, DI);
        wmma_gemm_tn<<<dim3((DI + 63) / 64, gy), blk128, 0, stream>>>(
            xdbl, XD, dtproj_w + (size_t)i * DI * RR, RR, dtl, DI, MM, DI, RR);
        scan_kernel<<<48, blk256, 0, stream>>>(
            xc, xdbl, dtl, dtproj_b + (size_t)i * DI,
            A_log + (size_t)i * DI * SS, D_param + (size_t)i * DI, ybuf, 0);

        // backward direction (first NSPA layers)
        if (i < NSPA) {
            conv_silu_kernel<<<(MM * DI + 255) / 256, blk256, 0, stream>>>(
                xz, conv_wb + (size_t)i * DI * 4, conv_bb + (size_t)i * DI, xc, 1);
            wmma_gemm_tn<<<dim3(1, gy), blk128, 0, stream>>>(
                xc, DI, xproj_wb + (size_t)i * XD * DI, DI, xdbl, XD, MM, XD, DI);
            wmma_gemm_tn<<<dim3((DI + 63) / 64, gy), blk128, 0, stream>>>(
                xdbl, XD, dtproj_wb + (size_t)i * DI * RR, RR, dtl, DI, MM, DI, RR);
            scan_kernel<<<48, blk256, 0, stream>>>(
                xc, xdbl, dtl, dtproj_bb + (size_t)i * DI,
                A_log_b + (size_t)i * DI * SS, D_b + (size_t)i * DI, ybuf, 1);
        }

        gate_kernel<<<(MM * DI + 255) / 256, blk256, 0, stream>>>(ybuf, xz);

        wmma_gemm_tn<<<dim3((EE + 63) / 64, gy), blk128, 0, stream>>>(
            ybuf, DI, outproj_w + (size_t)i * EE * DI, DI, hbuf, EE, MM, EE, DI);
    }

    // ---- final residual + rmsnorm ----
    resnorm_kernel<<<MM, blk128, 0, stream>>>(hbuf, res, norm_f_w, (float*)d_out, 0);
}